// LearnedPatches_63075889709170
// MI455X (gfx1250) — compile-verified
//
#include <hip/hip_runtime.h>

// out[32, 8192] = X[32, 16384] @ W[16384, 8192]   (all fp32, row-major)

typedef __attribute__((ext_vector_type(2))) float v2f;
typedef __attribute__((ext_vector_type(8))) float v8f;
typedef __attribute__((ext_vector_type(4))) int   v4i;

#define AS1 __attribute__((address_space(1)))
#define AS3 __attribute__((address_space(3)))

#if __has_builtin(__builtin_amdgcn_global_load_async_to_lds_b128)
#define HAVE_ASYNC_LDS 1
#endif

#define M_DIM   32
#define K_DIM   16384
#define N_DIM   8192
#define NTILE   128
#define KC      64
#define KSPLIT  8
#define NCHUNK  ((K_DIM / KSPLIT) / KC)   // 32 chunks per workgroup
#define SB_STRIDE (NTILE + 8)  // 136: rows 2 apart => +16 banks, conflict-free half-waves
#define SA_STRIDE (KC + 4)     // 68: 16B-aligned rows; 4*lane bank stride, halves disjoint

__device__ __forceinline__ void stage16(const float* __restrict__ g, float* l) {
#ifdef HAVE_ASYNC_LDS
    // GLOBAL_LOAD_ASYNC_TO_LDS_B128: LDS written directly, tracked by ASYNCcnt
    __builtin_amdgcn_global_load_async_to_lds_b128((AS1 v4i*)g, (AS3 v4i*)l, 0, 0);
#else
    *(float4*)l = *(const float4*)g;
#endif
}

__device__ __forceinline__ void wait_async_all() {
#ifdef HAVE_ASYNC_LDS
#if __has_builtin(__builtin_amdgcn_s_wait_asynccnt)
    __builtin_amdgcn_s_wait_asynccnt(0);
#else
    asm volatile("s_wait_asynccnt 0" ::: "memory");
#endif
#endif
}

__global__ __launch_bounds__(256) void lp_zero_kernel(float* __restrict__ out, int n) {
    int i = blockIdx.x * blockDim.x + threadIdx.x;
    if (i < n) out[i] = 0.0f;
}

__global__ __launch_bounds__(256) void lp_gemm_kernel(const float* __restrict__ X,
                                                      const float* __restrict__ W,
                                                      float* __restrict__ out) {
    __shared__ float sB[2][KC * SB_STRIDE];     // weight tiles  KC x NTILE (double buffer)
    __shared__ float sA[2][M_DIM * SA_STRIDE];  // act    tiles  32 x KC    (double buffer)

    const int tid  = threadIdx.x;
    const int lane = tid & 31;
    const int wave = tid >> 5;                        // 0..7
    const int n0   = blockIdx.x * NTILE;              // N tile base
    const int kbeg = blockIdx.y * (K_DIM / KSPLIT);   // K split base

    const int nc    = lane & 15;                      // fragment column / A row
    const int khalf = (lane >> 4) << 1;               // 0 or 2
    const int nw    = wave << 4;                      // wave's 16-col slice

    // staging coordinates (uniform across chunks)
    const int wrow  = tid >> 5;                       // + i*8, i = 0..7
    const int wcol4 = (tid & 31) << 2;                // 0..124 step 4
    const int arow  = tid >> 3;                       // 0..31
    const int acol4 = (tid & 7) << 2;                 // + j*32, j = 0..1

    v8f acc0 = {};  // output rows 0..15
    v8f acc1 = {};  // output rows 16..31

    auto stage = [&](int chunk, int buf) {
        const int kc = kbeg + chunk * KC;
        const float* gw = W + (size_t)kc * N_DIM + n0;
        float* lb = &sB[buf][0];
#pragma unroll
        for (int i = 0; i < 8; ++i) {                 // 64 x 128 weight tile
            const int r = wrow + (i << 3);
            stage16(gw + (size_t)r * N_DIM + wcol4, lb + r * SB_STRIDE + wcol4);
        }
        float* la = &sA[buf][0];
#pragma unroll
        for (int j = 0; j < 2; ++j) {                 // 32 x 64 activation tile
            const int c = acol4 + (j << 5);
            stage16(X + (size_t)arow * K_DIM + kc + c, la + arow * SA_STRIDE + c);
        }
    };

    stage(0, 0);                                      // prime the pipeline

    for (int ci = 0; ci < NCHUNK; ++ci) {
        wait_async_all();            // my async copies for chunk ci have landed
        __syncthreads();             // everyone's copies landed; prev compute done
        if (ci + 1 < NCHUNK) stage(ci + 1, (ci + 1) & 1);   // overlap with compute

        const float* lb = &sB[ci & 1][0];
        const float* la = &sA[ci & 1][0];
#pragma unroll
        for (int kk = 0; kk < KC; kk += 4) {          // 16 K-steps, 32 WMMAs
            const int kr = kk + khalf;
            v2f a0 = *(const v2f*)(la + nc * SA_STRIDE + kr);
            v2f a1 = *(const v2f*)(la + (16 + nc) * SA_STRIDE + kr);
            v2f b;
            b.x = lb[kr * SB_STRIDE + nw + nc];
            b.y = lb[(kr + 1) * SB_STRIDE + nw + nc];
            acc0 = __builtin_amdgcn_wmma_f32_16x16x4_f32(false, a0, false, b,
                                                         (short)0, acc0, false, false);
            acc1 = __builtin_amdgcn_wmma_f32_16x16x4_f32(false, a1, false, b,
                                                         (short)0, acc1, false, false);
        }
        // no trailing barrier needed: next iteration's top barrier protects buffers
    }

    // ---- epilogue: K-split partials accumulate via f32 atomics ----
    const int gn    = n0 + nw + nc;
    const int mhalf = (lane >> 4) << 3;   // 0 or 8
#pragma unroll
    for (int r = 0; r < 8; ++r) {
        atomicAdd(out + (size_t)(r + mhalf) * N_DIM + gn,      acc0[r]);
        atomicAdd(out + (size_t)(16 + r + mhalf) * N_DIM + gn, acc1[r]);
    }
}

extern "C" void kernel_launch(void* const* d_in, const int* in_sizes, int n_in,
                              void* d_out, int out_size, void* d_ws, size_t ws_size,
                              hipStream_t stream) {
    const float* X  = (const float*)d_in[0];   // (32, 128, 128)   -> (32, 16384)
    const float* W  = (const float*)d_in[1];   // (16384, 256, 32) -> (16384, 8192)
    float* out      = (float*)d_out;           // (32, 16, 16, 32) -> (32, 8192)

    lp_zero_kernel<<<(out_size + 255) / 256, 256, 0, stream>>>(out, out_size);

    dim3 grid(N_DIM / NTILE, KSPLIT);          // 64 x 8 = 512 workgroups
    lp_gemm_kernel<<<grid, 256, 0, stream>>>(X, W, out);
}